// Trifat_83494164234552
// MI455X (gfx1250) — compile-verified
//
#include <hip/hip_runtime.h>

// ---------------------------------------------------------------------------
// Trifat forward on MI455X (gfx1250, wave32).
// out = x @ Re(A)^T + bias : compose the linear operator once (phase 1, LDS),
// then run an fp32 WMMA GEMM 4096x1024x1024 with double-buffered LDS tiles.
// ---------------------------------------------------------------------------

#define NN    1024   // n
#define MM    10     // log2(n) butterfly stages per depth
#define NPERM 9      // m-1 block-perm stages per depth
#define TPB   256    // threads per transform block
#define EPT   4      // NN / TPB

typedef float v2f __attribute__((ext_vector_type(2)));
typedef float v8f __attribute__((ext_vector_type(8)));

__device__ __forceinline__ float2 cmul2(float2 a, float2 b) {
    return make_float2(fmaf(a.x, b.x, -(a.y * b.y)),
                       fmaf(a.x, b.y,  (a.y * b.x)));
}
__device__ __forceinline__ float2 cadd2(float2 a, float2 b) {
    return make_float2(a.x + b.x, a.y + b.y);
}
__device__ __forceinline__ float2 mix2(float p, float2 a, float2 b) {
    return make_float2(fmaf(p, b.x - a.x, a.x), fmaf(p, b.y - a.y, a.y));
}

// mode 0: input = identity column `row`, out[row][i] = Re(A)[i][row]  (=> Re(A)^T)
// mode 1: input = x[row][*], out[row][i] = Re(chain(x))[i] + bias[i]
__global__ __launch_bounds__(TPB) void trifat_transform(
    const float*  __restrict__ x,
    const float*  __restrict__ perm_logits,   // [depth][9][3]
    const float2* __restrict__ diags,         // [depth][10][1024]
    const float2* __restrict__ subdiags,
    const float2* __restrict__ superdiags,
    const float*  __restrict__ bias,
    float*        __restrict__ out,
    int depth, int mode)
{
    __shared__ float2 buf[2][NN];   // 16 KB ping-pong
    const int tid = threadIdx.x;
    const int row = blockIdx.x;
    int cur = 0;

    #pragma unroll
    for (int e = 0; e < EPT; ++e) {
        int i = e * TPB + tid;
        float re = (mode == 0) ? ((i == row) ? 1.0f : 0.0f)
                               : x[(size_t)row * NN + i];
        buf[0][i] = make_float2(re, 0.0f);
    }
    __syncthreads();

    for (int d = 0; d < depth; ++d) {
        for (int j = 0; j < NPERM; ++j) {
            const int s = 4 << j, h = s >> 1, smask = s - 1;
            const float* pl = perm_logits + ((size_t)d * NPERM + j) * 3;
            const float p0 = 1.0f / (1.0f + expf(-pl[0]));
            const float p1 = 1.0f / (1.0f + expf(-pl[1]));
            const float p2 = 1.0f / (1.0f + expf(-pl[2]));
            #pragma unroll
            for (int e = 0; e < EPT; ++e) {           // even-odd blend
                int i = e * TPB + tid;
                int q = i & smask, base = i - q;
                int partner = (q < h) ? (base + 2 * q) : (base + 2 * (q - h) + 1);
                buf[cur ^ 1][i] = mix2(p0, buf[cur][i], buf[cur][partner]);
            }
            __syncthreads(); cur ^= 1;
            #pragma unroll
            for (int e = 0; e < EPT; ++e) {           // half-reversal blends
                int i = e * TPB + tid;
                int q = i & smask, base = i - q;
                float pp; int mir;
                if (q < h) { pp = p1; mir = base + (h - 1 - q); }
                else       { pp = p2; mir = base + (s - 1 - q + h); }
                buf[cur ^ 1][i] = mix2(pp, buf[cur][i], buf[cur][mir]);
            }
            __syncthreads(); cur ^= 1;
        }
        for (int st = 0; st < MM; ++st) {
            const int k = NN >> (st + 1);
            const size_t wb = ((size_t)d * MM + st) * NN;
            #pragma unroll
            for (int e = 0; e < EPT; ++e) {
                int i = e * TPB + tid;
                float2 acc = cmul2(diags[wb + i], buf[cur][i]);
                if (i >= k)      acc = cadd2(acc, cmul2(subdiags[wb + i - k], buf[cur][i - k]));
                if (i < NN - k)  acc = cadd2(acc, cmul2(superdiags[wb + i],   buf[cur][i + k]));
                buf[cur ^ 1][i] = acc;
            }
            __syncthreads(); cur ^= 1;
        }
    }

    #pragma unroll
    for (int e = 0; e < EPT; ++e) {
        int i = e * TPB + tid;
        float v = buf[cur][i].x;
        if (mode == 1) v += bias[i];
        out[(size_t)row * NN + i] = v;
    }
}

// ---------------------------------------------------------------------------
// out[M][1024] = X[M][1024] * W[1024][1024] + bias, fp32 WMMA 16x16x4.
// Block: 256 thr (8 waves), tile 128M x 128N, K-chunk 16, double-buffered LDS.
//   Xs: row-major, stride 20 floats (16B-aligned b128 stores; A-frag
//       ds_load_b64 conflict-free: 20*l mod 64 distinct, gap>=4).
//   Wt: transposed [col][k], stride 20: B-frag is ONE ds_load_b64.
// Fragment layouts per CDNA5 ISA 7.12.2 (A 16x4 / B 4x16 / C 16x16).
// ---------------------------------------------------------------------------
#define GM  128
#define GN  128
#define KC  16
#define LSTR (KC + 4)          // 20 floats: LDS row/col stride

__global__ __launch_bounds__(256) void gemm_wmma_f32(
    const float* __restrict__ X, const float* __restrict__ W,
    const float* __restrict__ bias, float* __restrict__ out, int M)
{
    constexpr int N = NN, K = NN;
    __shared__ float Xs[2][GM * LSTR];   // 2 x 10 KB
    __shared__ float Wt[2][GN * LSTR];   // 2 x 10 KB

    const int t    = threadIdx.x;
    const int wave = t >> 5, lane = t & 31;
    const int half = lane >> 4, l15 = lane & 15;
    const int mBlk = blockIdx.y * GM;
    const int nBlk = blockIdx.x * GN;
    const int mw   = (wave & 3) * 32;    // wave M offset inside tile
    const int nw   = (wave >> 2) * 64;   // wave N offset inside tile

    v8f acc[2][4] = {};

    // ---- stage chunk 0 ----
    #pragma unroll
    for (int p = 0; p < 2; ++p) {        // X: f -> row=f/4, col4=f%4
        int f = t + p * 256, row = f >> 2, c4 = f & 3;
        float4 v = *(const float4*)(X + (size_t)(mBlk + row) * K + c4 * 4);
        *(float4*)&Xs[0][row * LSTR + c4 * 4] = v;
    }
    #pragma unroll
    for (int p = 0; p < 2; ++p) {        // W: f -> col=f%128, r4=f/128 (transpose)
        int f = t + p * 256, col = f & 127, r4 = f >> 7;
        float4 v;
        v.x = W[(size_t)(r4 * 4 + 0) * N + nBlk + col];
        v.y = W[(size_t)(r4 * 4 + 1) * N + nBlk + col];
        v.z = W[(size_t)(r4 * 4 + 2) * N + nBlk + col];
        v.w = W[(size_t)(r4 * 4 + 3) * N + nBlk + col];
        *(float4*)&Wt[0][col * LSTR + r4 * 4] = v;
    }
    __syncthreads();

    constexpr int nChunks = K / KC;      // 64
    for (int c = 0; c < nChunks; ++c) {
        const int buf = c & 1;
        const bool more = (c + 1 < nChunks);
        float4 px[2], pw[2];
        if (more) {                      // prefetch chunk c+1 into registers
            const int k0 = (c + 1) * KC;
            #pragma unroll
            for (int p = 0; p < 2; ++p) {
                int f = t + p * 256, row = f >> 2, c4 = f & 3;
                px[p] = *(const float4*)(X + (size_t)(mBlk + row) * K + k0 + c4 * 4);
            }
            #pragma unroll
            for (int p = 0; p < 2; ++p) {
                int f = t + p * 256, col = f & 127, r4 = f >> 7;
                pw[p].x = W[(size_t)(k0 + r4 * 4 + 0) * N + nBlk + col];
                pw[p].y = W[(size_t)(k0 + r4 * 4 + 1) * N + nBlk + col];
                pw[p].z = W[(size_t)(k0 + r4 * 4 + 2) * N + nBlk + col];
                pw[p].w = W[(size_t)(k0 + r4 * 4 + 3) * N + nBlk + col];
            }
        }
        // ---- compute on chunk c (32 WMMAs/wave) while prefetch is in flight ----
        #pragma unroll
        for (int kk = 0; kk < KC; kk += 4) {
            v2f a0 = *(const v2f*)&Xs[buf][(mw + l15)      * LSTR + kk + 2 * half];
            v2f a1 = *(const v2f*)&Xs[buf][(mw + 16 + l15) * LSTR + kk + 2 * half];
            #pragma unroll
            for (int tt = 0; tt < 4; ++tt) {
                v2f b = *(const v2f*)&Wt[buf][(nw + tt * 16 + l15) * LSTR + kk + 2 * half];
                acc[0][tt] = __builtin_amdgcn_wmma_f32_16x16x4_f32(
                    false, a0, false, b, (short)0, acc[0][tt], false, false);
                acc[1][tt] = __builtin_amdgcn_wmma_f32_16x16x4_f32(
                    false, a1, false, b, (short)0, acc[1][tt], false, false);
            }
        }
        __syncthreads();
        if (more) {                      // fill the other buffer
            #pragma unroll
            for (int p = 0; p < 2; ++p) {
                int f = t + p * 256, row = f >> 2, c4 = f & 3;
                *(float4*)&Xs[buf ^ 1][row * LSTR + c4 * 4] = px[p];
            }
            #pragma unroll
            for (int p = 0; p < 2; ++p) {
                int f = t + p * 256, col = f & 127, r4 = f >> 7;
                *(float4*)&Wt[buf ^ 1][col * LSTR + r4 * 4] = pw[p];
            }
        }
        __syncthreads();
    }

    // ---- epilogue: D layout (vgpr r: M=r / r+8 by lane half), fuse bias ----
    #pragma unroll
    for (int s = 0; s < 2; ++s) {
        #pragma unroll
        for (int tt = 0; tt < 4; ++tt) {
            const int col = nBlk + nw + tt * 16 + l15;
            const float bv = bias[col];
            #pragma unroll
            for (int r = 0; r < 8; ++r) {
                const int row = mBlk + mw + s * 16 + r + half * 8;
                out[(size_t)row * N + col] = acc[s][tt][r] + bv;
            }
        }
    }
}

extern "C" void kernel_launch(void* const* d_in, const int* in_sizes, int n_in,
                              void* d_out, int out_size, void* d_ws, size_t ws_size,
                              hipStream_t stream) {
    const float*  x     = (const float*)d_in[0];
    const float*  plog  = (const float*)d_in[1];
    const float2* diags = (const float2*)d_in[2];
    const float2* subd  = (const float2*)d_in[3];
    const float2* supd  = (const float2*)d_in[4];
    const float*  bias  = (const float*)d_in[5];
    float* outp = (float*)d_out;

    const int B     = in_sizes[0] / NN;                // 4096
    const int depth = in_sizes[2] / (MM * NN * 2);     // 2

    const size_t w_bytes = (size_t)NN * NN * sizeof(float);
    const bool gemm_path = (ws_size >= w_bytes) && (B % GM == 0);

    if (gemm_path) {
        float* W = (float*)d_ws;                       // W[j][i] = Re(A)[i][j]
        trifat_transform<<<NN, TPB, 0, stream>>>(
            nullptr, plog, diags, subd, supd, nullptr, W, depth, /*mode=*/0);
        dim3 grid(NN / GN, B / GM);                    // (8, 32)
        gemm_wmma_f32<<<grid, 256, 0, stream>>>(x, W, bias, outp, B);
    } else {
        trifat_transform<<<B, TPB, 0, stream>>>(
            x, plog, diags, subd, supd, bias, outp, depth, /*mode=*/1);
    }
}